// MaxKGINConv_62388694942256
// MI455X (gfx1250) — compile-verified
//
#include <hip/hip_runtime.h>

typedef float v2f __attribute__((ext_vector_type(2)));
typedef float v8f __attribute__((ext_vector_type(8)));

#define DFEAT 64
#define KTOP  32

// ---------------------------------------------------------------------------
// Kernel 1: reconstruct dense [N, 64] from top-k compressed features.
// One wave per node: zero a 64-float LDS row, scatter K=32 values with LDS
// float atomics (duplicates accumulate, matching .at[].add), write row out
// as coalesced float2.
// ---------------------------------------------------------------------------
__global__ __launch_bounds__(256) void maxk_scatter_kernel(
    const float* __restrict__ topk_values,
    const int*   __restrict__ topk_indices,
    float*       __restrict__ dense,
    int n_nodes) {
  __shared__ float rows[8 * DFEAT];             // 8 waves per 256-thread block
  const int wave_in_blk = threadIdx.x >> 5;
  const int lane        = threadIdx.x & 31;
  const int node        = blockIdx.x * 8 + wave_in_blk;
  float* row = &rows[wave_in_blk * DFEAT];

  row[lane]      = 0.0f;
  row[lane + 32] = 0.0f;
  __syncthreads();

  if (node < n_nodes) {
    const float v   = topk_values[(size_t)node * KTOP + lane];
    const int   idx = topk_indices[(size_t)node * KTOP + lane];
    atomicAdd(&row[idx], v);                    // ds_add_f32
  }
  __syncthreads();

  if (node < n_nodes) {
    float2 o;
    o.x = row[lane * 2];
    o.y = row[lane * 2 + 1];
    ((float2*)dense)[(size_t)node * (DFEAT / 2) + lane] = o;
  }
}

// ---------------------------------------------------------------------------
// Kernel 2: CSR segment-sum via V_WMMA_F32_16X16X4_F32 + GIN update.
// One wave owns 16 consecutive dst nodes; sorted edge_dst makes their edge
// set the contiguous range [lower_bound(base), lower_bound(base+16)).
// Per 4-edge chunk: A(16x4) one-hot dst-selection, B(4x16) gathered source
// feature tile, C accumulates 16x16. 4 tiles cover D=64. All loads are
// unconditional (index clamped); validity folds into the one-hot select so
// the loop body is pure load-clauses + WMMA (no divergent branches).
//
// Register layouts (ISA 7.12.2, wave32):
//   A 16x4 f32 : lane L holds M = L%16 ; VGPR j -> K = 2*(L/16) + j
//   B 4x16 f32 : lane L holds N = L%16 ; VGPR j -> K = 2*(L/16) + j
//   C/D 16x16  : lane L holds N = L%16 ; VGPR r -> M = r + 8*(L/16)
// ---------------------------------------------------------------------------
__global__ __launch_bounds__(256) void spmm_wmma_gin_kernel(
    const float* __restrict__ dense,
    const float* __restrict__ feat,
    const float* __restrict__ eps,
    const int*   __restrict__ edge_src,
    const int*   __restrict__ edge_dst,
    float*       __restrict__ out,
    int n_nodes, int n_edges) {
  const int lane = threadIdx.x & 31;
  const int wave = (blockIdx.x * blockDim.x + threadIdx.x) >> 5;
  const int base = wave * 16;                    // first dst node of group
  if (base >= n_nodes) return;                   // wave-uniform: EXEC stays all-1

  // lower_bound over sorted edge_dst (uniform across the wave)
  int e0, e1;
  {
    int lo = 0, hi = n_edges;
    while (lo < hi) { int mid = (lo + hi) >> 1;
                      if (edge_dst[mid] < base) lo = mid + 1; else hi = mid; }
    e0 = lo;
    hi = n_edges;
    while (lo < hi) { int mid = (lo + hi) >> 1;
                      if (edge_dst[mid] < base + 16) lo = mid + 1; else hi = mid; }
    e1 = lo;
  }

  const int half = lane >> 4;                    // 0 or 1
  const int m    = lane & 15;                    // M for A, N for B/C
  const int tgt  = base + m;                     // dst id this lane selects
  const int kofs = half * 2;                     // K pair this lane supplies
  const int last = n_edges - 1;
  v8f acc[4] = {};                               // feature tiles n0 = 0,16,32,48

  for (int c = e0; c < e1; c += 8) {             // 2 chunks of 4 edges per iter
    // ---- gather edge ids for both chunks (one clause, clamped indices) ----
    const int pA0 = c + kofs,     pB0 = pA0 + 1;
    const int pA1 = c + 4 + kofs, pB1 = pA1 + 1;
    const int eA0 = min(pA0, last), eB0 = min(pB0, last);
    const int eA1 = min(pA1, last), eB1 = min(pB1, last);
    const int dA0 = edge_dst[eA0], dB0 = edge_dst[eB0];
    const int dA1 = edge_dst[eA1], dB1 = edge_dst[eB1];
    const int sA0 = edge_src[eA0], sB0 = edge_src[eB0];
    const int sA1 = edge_src[eA1], sB1 = edge_src[eB1];

    v2f a0, a1;                                  // one-hot selection columns
    a0.x = (pA0 < e1 && dA0 == tgt) ? 1.0f : 0.0f;
    a0.y = (pB0 < e1 && dB0 == tgt) ? 1.0f : 0.0f;
    a1.x = (pA1 < e1 && dA1 == tgt) ? 1.0f : 0.0f;
    a1.y = (pB1 < e1 && dB1 == tgt) ? 1.0f : 0.0f;

    const float* rA0 = dense + (size_t)sA0 * DFEAT;
    const float* rB0 = dense + (size_t)sB0 * DFEAT;
    const float* rA1 = dense + (size_t)sA1 * DFEAT;
    const float* rB1 = dense + (size_t)sB1 * DFEAT;

    // ---- gather B tiles for both chunks before any WMMA (max MLP) ----
    v2f b0[4], b1[4];
#pragma unroll
    for (int t = 0; t < 4; ++t) {
      b0[t].x = rA0[t * 16 + m];
      b0[t].y = rB0[t * 16 + m];
      b1[t].x = rA1[t * 16 + m];
      b1[t].y = rB1[t * 16 + m];
    }

#pragma unroll
    for (int t = 0; t < 4; ++t)                  // chunk 0: 4 independent tiles
      acc[t] = __builtin_amdgcn_wmma_f32_16x16x4_f32(
          false, a0, false, b0[t], (short)0, acc[t], false, false);
#pragma unroll
    for (int t = 0; t < 4; ++t)                  // chunk 1
      acc[t] = __builtin_amdgcn_wmma_f32_16x16x4_f32(
          false, a1, false, b1[t], (short)0, acc[t], false, false);
  }

  const float ep = 1.0f + eps[0];
#pragma unroll
  for (int t = 0; t < 4; ++t) {
#pragma unroll
    for (int r = 0; r < 8; ++r) {
      const int row = base + r + 8 * half;       // M of this C/D element
      const int col = t * 16 + m;                // N of this C/D element
      const size_t o = (size_t)row * DFEAT + col;
      out[o] = ep * feat[o] + acc[t][r];
    }
  }
}

// ---------------------------------------------------------------------------
extern "C" void kernel_launch(void* const* d_in, const int* in_sizes, int n_in,
                              void* d_out, int out_size, void* d_ws, size_t ws_size,
                              hipStream_t stream) {
  const float* feat         = (const float*)d_in[0];
  const float* topk_values  = (const float*)d_in[1];
  const float* eps          = (const float*)d_in[2];
  const int*   topk_indices = (const int*)  d_in[3];
  const int*   edge_src     = (const int*)  d_in[4];
  const int*   edge_dst     = (const int*)  d_in[5];
  float*       out          = (float*)d_out;

  const int n_nodes = in_sizes[0] / DFEAT;
  const int n_edges = in_sizes[4];

  float* dense = (float*)d_ws;                   // N*64 f32 = 16 MB scratch

  // Stage 1: dense reconstruction (8 nodes per 256-thread block)
  {
    const int blocks = (n_nodes + 7) / 8;
    maxk_scatter_kernel<<<blocks, 256, 0, stream>>>(
        topk_values, topk_indices, dense, n_nodes);
  }

  // Stage 2: WMMA segment-sum + GIN update (one wave per 16 dst nodes)
  {
    const int waves  = (n_nodes + 15) / 16;
    const int blocks = (waves + 7) / 8;          // 8 waves per block
    spmm_wmma_gin_kernel<<<blocks, 256, 0, stream>>>(
        dense, feat, eps, edge_src, edge_dst, out, n_nodes, n_edges);
  }
}